// GCNNet_70531952934932
// MI455X (gfx1250) — compile-verified
//
#include <hip/hip_runtime.h>
#include <hip/hip_bf16.h>

#define HDIM 128
#define CAP 64          // adjacency bucket capacity per node (mean in-degree = 16)
#define MAX_OVF 8192    // bounded overflow list (statistically never used)

typedef float v2f __attribute__((ext_vector_type(2)));
typedef float v8f __attribute__((ext_vector_type(8)));

// Fused: out = LayerNorm( relu(X @ Wm + bias) + R ) * g + bln
// One wave computes a 16-row x 128-col strip via V_WMMA_F32_16X16X4_F32.
__global__ __launch_bounds__(256, 1)
void gemm_bias_relu_res_ln(const float* __restrict__ X,
                           const float* __restrict__ Wm,
                           const float* __restrict__ bias,
                           const float* __restrict__ R,
                           const float* __restrict__ g,
                           const float* __restrict__ bln,
                           float* __restrict__ out,
                           int n_rows)
{
    __shared__ float Ws[HDIM * HDIM];   // 64 KB: whole weight matrix
    const int tid = threadIdx.x;

    // Stage W into LDS (float4 vectorized: 16384 floats / 256 threads)
    {
        const float4* Wv = (const float4*)Wm;
        float4* Sv = (float4*)Ws;
#pragma unroll
        for (int i = 0; i < (HDIM * HDIM / 4) / 256; ++i)
            Sv[tid + i * 256] = Wv[tid + i * 256];
    }
    __syncthreads();

    const int wave = tid >> 5;
    const int lane = tid & 31;
    const int half = lane >> 4;   // 0: lanes 0-15, 1: lanes 16-31
    const int l16  = lane & 15;

    const int row_base = blockIdx.x * 128 + wave * 16;

    // A-matrix 16x4 layout: lanes 0-15 hold K=0,1 ; lanes 16-31 hold K=2,3
    int a_row = row_base + l16;
    if (a_row > n_rows - 1) a_row = n_rows - 1;   // clamp; results discarded
    const float* Xrow = X + (size_t)a_row * HDIM;

    v8f acc[8];
#pragma unroll
    for (int nt = 0; nt < 8; ++nt)
        acc[nt] = v8f{0.f, 0.f, 0.f, 0.f, 0.f, 0.f, 0.f, 0.f};

    // K loop: 32 steps of K=4
#pragma unroll 4
    for (int ks = 0; ks < 32; ++ks) {
        const int kb = ks * 4 + half * 2;
        v2f a;
        a.x = Xrow[kb];
        a.y = Xrow[kb + 1];
#pragma unroll
        for (int nt = 0; nt < 8; ++nt) {
            // B-matrix 4x16: lanes 0-15 hold K=0,1 ; lanes 16-31 hold K=2,3
            const int n = nt * 16 + l16;
            v2f bf;
            bf.x = Ws[kb * HDIM + n];
            bf.y = Ws[(kb + 1) * HDIM + n];
            acc[nt] = __builtin_amdgcn_wmma_f32_16x16x4_f32(
                false, a, false, bf, (short)0, acc[nt], false, false);
        }
    }

    // Per-column parameters (column n = nt*16 + l16)
    float biasv[8], gv[8], bv[8];
#pragma unroll
    for (int nt = 0; nt < 8; ++nt) {
        const int n = nt * 16 + l16;
        biasv[nt] = bias[n];
        gv[nt]    = g[n];
        bv[nt]    = bln[n];
    }

    // Epilogue. C/D layout: VGPR j, lanes 0-15 -> row j, lanes 16-31 -> row j+8,
    // column = nt*16 + l16.
#pragma unroll
    for (int j = 0; j < 8; ++j) {
        const int r  = row_base + j + half * 8;
        const int rc = (r > n_rows - 1) ? (n_rows - 1) : r;
        const float* Rrow = R + (size_t)rc * HDIM;

        float s = 0.f;
#pragma unroll
        for (int nt = 0; nt < 8; ++nt) {
            float x = acc[nt][j] + biasv[nt];
            x = fmaxf(x, 0.f);                 // relu
            x += Rrow[nt * 16 + l16];          // residual
            acc[nt][j] = x;
            s += x;
        }
        // Half-wave reduction: masks <= 8 keep lanes 0-15 / 16-31 separate
        s += __shfl_xor(s, 1, 32);
        s += __shfl_xor(s, 2, 32);
        s += __shfl_xor(s, 4, 32);
        s += __shfl_xor(s, 8, 32);
        const float mean = s * (1.f / HDIM);

        float var = 0.f;
#pragma unroll
        for (int nt = 0; nt < 8; ++nt) {
            const float d = acc[nt][j] - mean;
            var += d * d;
        }
        var += __shfl_xor(var, 1, 32);
        var += __shfl_xor(var, 2, 32);
        var += __shfl_xor(var, 4, 32);
        var += __shfl_xor(var, 8, 32);
        const float inv = rsqrtf(var * (1.f / HDIM) + 1e-5f);

#pragma unroll
        for (int nt = 0; nt < 8; ++nt)
            acc[nt][j] = (acc[nt][j] - mean) * inv * gv[nt] + bv[nt];

        if (r < n_rows) {
            float* orow = out + (size_t)r * HDIM;
#pragma unroll
            for (int nt = 0; nt < 8; ++nt)
                orow[nt * 16 + l16] = acc[nt][j];
        }
    }
}

// Build fixed-capacity dst-bucketed adjacency (counting pass, int atomics only).
__global__ __launch_bounds__(256)
void build_adj(const int* __restrict__ src, const int* __restrict__ dst,
               int* __restrict__ cnt, int* __restrict__ adj,
               int* __restrict__ ovf_cnt, int* __restrict__ ovf, int n_edges)
{
    const int e = blockIdx.x * 256 + threadIdx.x;
    if (e >= n_edges) return;
    const int d = dst[e];
    const int p = atomicAdd(&cnt[d], 1);
    if (p < CAP) {
        adj[(size_t)d * CAP + p] = src[e];
    } else {
        const int o = atomicAdd(ovf_cnt, 1);
        if (o < MAX_OVF) ovf[o] = e;
    }
}

// Atomic-free aggregation: one wave per node gathers + sums its in-neighbors.
// Fully overwrites agg (no memset needed).
__global__ __launch_bounds__(256)
void gather_agg(const float* __restrict__ ff, const int* __restrict__ adj,
                const int* __restrict__ cnt, float* __restrict__ agg, int n_nodes)
{
    const int gid = blockIdx.x * 256 + threadIdx.x;
    const int n = gid >> 5;
    if (n >= n_nodes) return;
    const int lane = threadIdx.x & 31;

    int deg = cnt[n];
    if (deg > CAP) deg = CAP;
    const int* ap = adj + (size_t)n * CAP;

    float4 s = {0.f, 0.f, 0.f, 0.f};
    for (int i = 0; i < deg; ++i) {
        const int sn = ap[i];
        const float4 v = *(const float4*)(ff + (size_t)sn * HDIM + lane * 4);
        s.x += v.x; s.y += v.y; s.z += v.z; s.w += v.w;
    }
    *(float4*)(agg + (size_t)n * HDIM + lane * 4) = s;
}

// Correctness fallback: atomically add any bucket-overflow edges (normally zero).
__global__ __launch_bounds__(256)
void ovf_scatter(const float* __restrict__ ff, const int* __restrict__ src,
                 const int* __restrict__ dst, const int* __restrict__ ovf,
                 const int* __restrict__ ovf_cnt, float* __restrict__ agg)
{
    const int gid = blockIdx.x * 256 + threadIdx.x;
    const int idx = gid >> 5;
    int c = *ovf_cnt;
    if (c > MAX_OVF) c = MAX_OVF;
    if (idx >= c) return;
    const int lane = threadIdx.x & 31;

    const int e = ovf[idx];
    const int s = src[e];
    const int d = dst[e];
    const float4 v = *(const float4*)(ff + (size_t)s * HDIM + lane * 4);
    float* ap = agg + (size_t)d * HDIM + lane * 4;
    unsafeAtomicAdd(ap + 0, v.x);
    unsafeAtomicAdd(ap + 1, v.y);
    unsafeAtomicAdd(ap + 2, v.z);
    unsafeAtomicAdd(ap + 3, v.w);
}

extern "C" void kernel_launch(void* const* d_in, const int* in_sizes, int n_in,
                              void* d_out, int out_size, void* d_ws, size_t ws_size,
                              hipStream_t stream)
{
    const float* features = (const float*)d_in[0];
    const int*   src      = (const int*)d_in[1];
    const int*   dst      = (const int*)d_in[2];
    const float* W        = (const float*)d_in[3];
    const float* b        = (const float*)d_in[4];
    const float* gcn_W    = (const float*)d_in[5];
    const float* gcn_b    = (const float*)d_in[6];
    const float* ffln_g   = (const float*)d_in[7];
    const float* ffln_b   = (const float*)d_in[8];
    const float* ln_g     = (const float*)d_in[9];
    const float* ln_b     = (const float*)d_in[10];

    const int n_nodes  = in_sizes[0] / HDIM;
    const int n_edges  = in_sizes[1];
    const int n_layers = in_sizes[6] / HDIM;   // gcn_b is (NLAYERS, HDIM)

    // Workspace layout
    char* ws = (char*)d_ws;
    float* ff      = (float*)ws;                                       // 25.6 MB
    float* agg     = ff + (size_t)n_nodes * HDIM;                      // 25.6 MB
    int*   adj     = (int*)(agg + (size_t)n_nodes * HDIM);             // 12.8 MB
    int*   cnt     = adj + (size_t)n_nodes * CAP;                      // 200 KB
    int*   ovf_cnt = cnt + n_nodes;                                    // 4 B
    int*   ovf     = ovf_cnt + 1;                                      // 32 KB
    float* out     = (float*)d_out;

    const dim3 block(256);
    const dim3 ggrid((n_nodes + 127) / 128);                           // gemm blocks
    const dim3 egrid((n_edges + 255) / 256);                           // per-edge threads
    const dim3 ngrid((unsigned)(((size_t)n_nodes * 32 + 255) / 256));  // wave per node
    const dim3 ogrid((MAX_OVF * 32) / 256);                            // overflow waves

    // Build dst-bucketed adjacency once per call (cnt + ovf_cnt zeroed together)
    hipMemsetAsync(cnt, 0, ((size_t)n_nodes + 1) * sizeof(int), stream);
    build_adj<<<egrid, block, 0, stream>>>(src, dst, cnt, adj, ovf_cnt, ovf, n_edges);

    const float* cur = features;
    for (int i = 0; i < n_layers; ++i) {
        // ff = LN(relu(cur @ W + b) + cur)
        gemm_bias_relu_res_ln<<<ggrid, block, 0, stream>>>(
            cur, W, b, cur, ffln_g + i * HDIM, ffln_b + i * HDIM, ff, n_nodes);

        // agg = segment_sum(ff[src], dst)  — atomic-free gather
        gather_agg<<<ngrid, block, 0, stream>>>(ff, adj, cnt, agg, n_nodes);
        ovf_scatter<<<ogrid, block, 0, stream>>>(ff, src, dst, ovf, ovf_cnt, agg);

        // out = LN(relu(agg @ gcn_W[i] + gcn_b[i]) + ff)
        gemm_bias_relu_res_ln<<<ggrid, block, 0, stream>>>(
            agg, gcn_W + (size_t)i * HDIM * HDIM, gcn_b + i * HDIM,
            ff, ln_g + i * HDIM, ln_b + i * HDIM, out, n_nodes);

        cur = out;   // layer 1 reads d_out, rewrites d_out
    }
}